// CumprodOneMinusAlphaToTransmittanceModule_89790586290715
// MI455X (gfx1250) — compile-verified
//
#include <hip/hip_runtime.h>
#include <math.h>

#define TOTAL_ELEMS 16777216
#define NUM_RAYS    262144
#define EPS_F       1e-6f
#define NTILES      8192      // TOTAL / 2048
#define TILE_ELEMS  2048      // 8 waves * 256 elems

#define LN2_F    0.6931471805599453f
#define LOG2E_F  1.4426950408889634f

typedef __attribute__((ext_vector_type(2))) float v2f;
typedef __attribute__((ext_vector_type(8))) float v8f;

// log1p(-clip(a,0,1-eps)) on the TRANS unit.
// u = 1-a in [1e-6, 1] (no denorm/inf/NaN): ln(u) = v_log_f32(u)*ln2.
// Kahan correction: log1p(-a) = ln(u) * (-a)/(u-1); exact -a when u rounds to 1.
__device__ __forceinline__ float lg_of_alpha(float a) {
    a = fminf(fmaxf(a, 0.0f), 1.0f - EPS_F);
    const float u = 1.0f - a;
    const float d = u - 1.0f;                       // == -a up to the rounding in u
    const float r = __builtin_amdgcn_logf(u) * (LN2_F * (-a)) * __builtin_amdgcn_rcpf(d);
    return (d == 0.0f) ? -a : r;
}

// exp(x) for x <= 0 (underflow to 0 is desired): v_exp_f32(x*log2e)
__device__ __forceinline__ float fast_exp(float x) {
    return __builtin_amdgcn_exp2f(x * LOG2E_F);
}

__device__ __forceinline__ float lane_bperm(int srcLane, float v) {
    return __int_as_float(__builtin_amdgcn_ds_bpermute(srcLane << 2, __float_as_int(v)));
}

// ---------------- K1: per-tile sum of lg ----------------
__global__ void k_tile_reduce(const float* __restrict__ alpha,
                              float* __restrict__ tileSums) {
    __shared__ float wsum[8];
    const int tile = blockIdx.x;
    const int t    = threadIdx.x;           // 256 threads
    const float4* p = (const float4*)(alpha + (size_t)tile * TILE_ELEMS + t * 8);
    float4 a0 = p[0], a1 = p[1];
    float s = lg_of_alpha(a0.x) + lg_of_alpha(a0.y) + lg_of_alpha(a0.z) + lg_of_alpha(a0.w)
            + lg_of_alpha(a1.x) + lg_of_alpha(a1.y) + lg_of_alpha(a1.z) + lg_of_alpha(a1.w);
    #pragma unroll
    for (int off = 16; off >= 1; off >>= 1) s += __shfl_xor(s, off, 32);
    if ((t & 31) == 0) wsum[t >> 5] = s;
    __syncthreads();
    if (t == 0) {
        float tot = 0.f;
        #pragma unroll
        for (int w = 0; w < 8; ++w) tot += wsum[w];
        tileSums[tile] = tot;
    }
}

// ---------------- K2: exclusive scan of 8192 tile sums (in place) ----------------
__global__ void k_scan_tiles(float* __restrict__ tileSums, float* __restrict__ P) {
    __shared__ float wtot[32];
    const int t = threadIdx.x;              // 1024 threads
    const int lane = t & 31, wid = t >> 5;
    float v[8];
    float sum = 0.f;
    #pragma unroll
    for (int j = 0; j < 8; ++j) { v[j] = tileSums[t * 8 + j]; sum += v[j]; }
    float incl = sum;
    #pragma unroll
    for (int off = 1; off < 32; off <<= 1) {
        float tmp = lane_bperm((lane - off) & 31, incl);
        if (lane >= off) incl += tmp;
    }
    if (lane == 31) wtot[wid] = incl;
    __syncthreads();
    if (wid == 0) {
        float xi = wtot[lane];
        #pragma unroll
        for (int off = 1; off < 32; off <<= 1) {
            float tmp = lane_bperm((lane - off) & 31, xi);
            if (lane >= off) xi += tmp;
        }
        wtot[lane] = xi;                    // inclusive wave totals
    }
    __syncthreads();
    const float waveOff = (wid == 0) ? 0.f : wtot[wid - 1];
    float run = waveOff + incl - sum;       // thread-exclusive base
    #pragma unroll
    for (int j = 0; j < 8; ++j) { float tmp = v[j]; tileSums[t * 8 + j] = run; run += tmp; }
    if (t == 0) P[0] = 0.f;
}

// ---------------- K3: WMMA tile scan, writes P[e+1] = cs[e] ----------------
// Per wave: 256 elems as X[16][16] column-major (elem = n*16 + k).
// D = L * X via 4 chained V_WMMA_F32_16X16X4_F32 (L = lower-triangular ones).
__global__ void k_scan_main(const float* __restrict__ alpha,
                            const float* __restrict__ tileOff,
                            float* __restrict__ P) {
    __shared__ float waveTot[8];
    const int tile = blockIdx.x;
    const int t    = threadIdx.x;           // 256 threads = 8 waves
    const int wid  = t >> 5, lane = t & 31;
    const int n    = lane & 15;             // column / M index
    const int hi   = lane >> 4;             // half-wave selector
    const size_t chunkBase = (size_t)tile * TILE_ELEMS + (size_t)wid * 256;

    // Load B panels (ISA layout: VGPR0 rows {4p+0 | 4p+2}, VGPR1 rows {4p+1 | 4p+3})
    v2f b[4];
    #pragma unroll
    for (int p = 0; p < 4; ++p) {
        const int k = p * 4 + hi * 2;
        const float2 a2 = *(const float2*)(alpha + chunkBase + (size_t)n * 16 + k);
        b[p].x = lg_of_alpha(a2.x);
        b[p].y = lg_of_alpha(a2.y);
    }
    v8f c = {0.f,0.f,0.f,0.f,0.f,0.f,0.f,0.f};
    #pragma unroll
    for (int p = 0; p < 4; ++p) {
        const int k = p * 4 + hi * 2;       // A layout: lanes 0-15 K=0,1; 16-31 K=2,3
        v2f a;
        a.x = (k     <= n) ? 1.f : 0.f;     // inclusive lower-triangular ones
        a.y = (k + 1 <= n) ? 1.f : 0.f;
        c = __builtin_amdgcn_wmma_f32_16x16x4_f32(false, a, false, b[p],
                                                  (short)0, c, false, false);
    }
    // Cross-column stitch: colsum[n] = D[15][n] (c[7] of lanes 16-31)
    float colsum = lane_bperm(16 + n, c[7]);
    float incl = colsum;
    #pragma unroll
    for (int off = 1; off < 16; off <<= 1) {
        float tmp = lane_bperm((lane - off) & 31, incl);   // wraps; masked below
        if (n >= off) incl += tmp;
    }
    const float coloff   = incl - colsum;   // exclusive column offset
    const float chunkTot = __int_as_float(
        __builtin_amdgcn_readlane(__float_as_int(incl), 15));
    if (lane == 0) waveTot[wid] = chunkTot;
    __syncthreads();
    float waveBase = 0.f;
    for (int w = 0; w < wid; ++w) waveBase += waveTot[w];
    const float base = tileOff[tile] + waveBase + coloff;

    // C/D layout: c[v] = D[v + 8*hi][n]; element e = n*16 + row. Store P[e+1]=cs[e].
    #pragma unroll
    for (int v = 0; v < 8; ++v) {
        const int row = v + hi * 8;
        const size_t e = chunkBase + (size_t)n * 16 + row;
        P[e + 1] = c[v] + base;
    }
}

// ---------------- K5: per-ray base + background transmittance ----------------
__global__ void k_rays(const int2* __restrict__ se, const float* __restrict__ P,
                       float* __restrict__ baseArr, float* __restrict__ bg) {
    const int r = blockIdx.x * blockDim.x + threadIdx.x;
    if (r >= NUM_RAYS) return;
    const int2 v = se[r];
    const float b = P[v.x];                 // cs_excl[start]  (P has TOTAL+1 entries)
    baseArr[r] = b;
    float out = 1.0f;
    if (v.y > v.x) out = fast_exp(P[v.y] - b);  // cs[end-1] == P[end]
    bg[r] = out;
}

// ---------------- K4: per-sample transmittance ----------------
__global__ void k_transmit(const float* __restrict__ P, const int* __restrict__ ray_id,
                           const float* __restrict__ baseArr, float* __restrict__ out) {
    const int idx = (blockIdx.x * blockDim.x + threadIdx.x) * 4;
    const float4 p = *(const float4*)(P + idx);        // cs_excl[i..i+3]
    const int4   r = *(const int4*)(ray_id + idx);
    float4 o;
    o.x = fast_exp(p.x - baseArr[r.x]);
    o.y = fast_exp(p.y - baseArr[r.y]);
    o.z = fast_exp(p.z - baseArr[r.z]);
    o.w = fast_exp(p.w - baseArr[r.w]);
    *(float4*)(out + idx) = o;
}

extern "C" void kernel_launch(void* const* d_in, const int* in_sizes, int n_in,
                              void* d_out, int out_size, void* d_ws, size_t ws_size,
                              hipStream_t stream) {
    const float* alpha  = (const float*)d_in[0];
    const int*   se     = (const int*)d_in[1];   // (N_RAYS, 2) int32
    const int*   ray_id = (const int*)d_in[2];
    float* out = (float*)d_out;                  // [transmittance | bg] concatenated

    float* P       = (float*)d_ws;               // TOTAL+1 floats (padded to +4)
    float* tileBuf = P + (TOTAL_ELEMS + 4);      // NTILES floats
    float* baseArr = tileBuf + NTILES;           // NUM_RAYS floats

    k_tile_reduce<<<NTILES, 256, 0, stream>>>(alpha, tileBuf);
    k_scan_tiles <<<1, 1024, 0, stream>>>(tileBuf, P);
    k_scan_main  <<<NTILES, 256, 0, stream>>>(alpha, tileBuf, P);
    k_rays       <<<NUM_RAYS / 256, 256, 0, stream>>>((const int2*)se, P, baseArr,
                                                      out + TOTAL_ELEMS);
    k_transmit   <<<TOTAL_ELEMS / 1024, 256, 0, stream>>>(P, ray_id, baseArr, out);
}